// CapsuleLayer_13950053778028
// MI455X (gfx1250) — compile-verified
//
#include <hip/hip_runtime.h>
#include <hip/hip_bf16.h>

// ---------------- problem constants (from reference setup_inputs) ----------
#define BATCH 128
#define RNODES 1152
#define CI 8
#define CAPS 10
#define OUT 16
#define NITER 3
#define SOUT (CAPS * BATCH * OUT)   // 20480

typedef float v2f __attribute__((ext_vector_type(2)));
typedef float v8f __attribute__((ext_vector_type(8)));

// ---------------------------------------------------------------------------
// Kernel 0: zero Vsum (20480 floats) and the 3 norm accumulators.
// ---------------------------------------------------------------------------
__global__ void caps_init(float* __restrict__ Vsum, float* __restrict__ n2) {
    int i = blockIdx.x * blockDim.x + threadIdx.x;
    if (i < SOUT) Vsum[i] = 0.0f;
    if (i < NITER) n2[i] = 0.0f;
}

// ---------------------------------------------------------------------------
// Kernel 1: u[c][b][r][o] = sum_i x[b][r][i] * W[c][r][i][o]
// One wave per (c, r): B operand (8x16 W slice) loaded once, reused across
// 8 b-tiles; 2x V_WMMA_F32_16X16X4_F32 per b-tile (K = 0..3, 4..7).
// Grid: exactly C*R waves = 11520 -> 1440 blocks x 256 threads. EXEC all-1s.
// ---------------------------------------------------------------------------
__global__ void caps_gemm(const float* __restrict__ x,
                          const float* __restrict__ W,
                          float* __restrict__ u) {
    const int wave = blockIdx.x * 8 + (threadIdx.x >> 5);  // 0 .. 11519
    const int c = wave / RNODES;
    const int r = wave % RNODES;
    const int lane = threadIdx.x & 31;
    const int half = lane >> 4;        // 0: K=k0..k0+1, 1: K=k0+2..k0+3
    const int m    = lane & 15;        // A row / B col / D col
    const int k0   = half * 2;

    // B operand: W[c][r][k][n], n = m.  b1 covers K=0..3, b2 covers K=4..7.
    const long wbase = ((long)(c * RNODES + r) * CI) * OUT + m;
    v2f b1, b2;
    b1.x = W[wbase + (long)(k0 + 0) * OUT];
    b1.y = W[wbase + (long)(k0 + 1) * OUT];
    b2.x = W[wbase + (long)(k0 + 4) * OUT];
    b2.y = W[wbase + (long)(k0 + 5) * OUT];

    #pragma unroll
    for (int bt = 0; bt < 8; ++bt) {
        const int b0 = bt * 16;
        // A operand: x[b0+m][r][k0..k0+1] and [k0+4..k0+5], 8B vector loads.
        const float* xp = x + ((long)(b0 + m) * RNODES + r) * CI;
        v2f a1 = *(const v2f*)(xp + k0);
        v2f a2 = *(const v2f*)(xp + k0 + 4);

        v8f acc = {};
        acc = __builtin_amdgcn_wmma_f32_16x16x4_f32(false, a1, false, b1,
                                                    (short)0, acc, false, false);
        acc = __builtin_amdgcn_wmma_f32_16x16x4_f32(false, a2, false, b2,
                                                    (short)0, acc, false, false);

        // D layout: VGPR j holds row M = j + 8*half, col N = m.
        // Store to u[c][b0+M][r][o=m]; 16 consecutive lanes -> 64B contiguous.
        #pragma unroll
        for (int j = 0; j < 8; ++j) {
            const int b = b0 + j + 8 * half;
            u[((long)(c * BATCH + b) * RNODES + r) * OUT + m] = acc[j];
        }
    }
}

// ---------------------------------------------------------------------------
// Kernel 2: one routing pass.  logits = u * Vsum (algebraic identity),
// s[c][b][o] = sum_r exp(u*Vs)*u / sum_r exp(u*Vs).  Iteration 0 (Vsum = 0)
// degenerates to the uniform softmax automatically.
// Grid: C*B = 1280 blocks x 256 threads; tid -> (rg = tid>>2 strides r,
// og = tid&3 owns 4 o's via float4).  Block accumulates sum(s^2) into n2.
// ---------------------------------------------------------------------------
__global__ void caps_route(const float* __restrict__ u,
                           const float* __restrict__ Vsum,
                           float* __restrict__ s,
                           float* __restrict__ n2) {
    const int cb = blockIdx.x;                      // c*BATCH + b
    const int og = threadIdx.x & 3;
    const int rg = threadIdx.x >> 2;                // 0..63
    const long ubase = (long)cb * RNODES * OUT + og * 4;

    const float4 vs = *(const float4*)(Vsum + cb * OUT + og * 4);

    float4 num = make_float4(0.f, 0.f, 0.f, 0.f);
    float4 den = make_float4(0.f, 0.f, 0.f, 0.f);

    for (int r = rg; r < RNODES; r += 64) {
        const float4 uv = *(const float4*)(u + ubase + (long)r * OUT);
        __builtin_prefetch(u + ubase + (long)(r + 64) * OUT, 0, 1);
        float4 e;
        e.x = __expf(uv.x * vs.x);
        e.y = __expf(uv.y * vs.y);
        e.z = __expf(uv.z * vs.z);
        e.w = __expf(uv.w * vs.w);
        num.x += e.x * uv.x;  num.y += e.y * uv.y;
        num.z += e.z * uv.z;  num.w += e.w * uv.w;
        den.x += e.x;  den.y += e.y;  den.z += e.z;  den.w += e.w;
    }

    __shared__ float4 numS[256];
    __shared__ float4 denS[256];
    numS[threadIdx.x] = num;
    denS[threadIdx.x] = den;
    __syncthreads();

    // reduce over rg (stride 4 in tid), keep og lanes independent
    #pragma unroll
    for (int h = 32; h >= 1; h >>= 1) {
        if (rg < h) {
            const int p = threadIdx.x + h * 4;
            float4 a = numS[threadIdx.x], bb = numS[p];
            a.x += bb.x; a.y += bb.y; a.z += bb.z; a.w += bb.w;
            numS[threadIdx.x] = a;
            a = denS[threadIdx.x]; bb = denS[p];
            a.x += bb.x; a.y += bb.y; a.z += bb.z; a.w += bb.w;
            denS[threadIdx.x] = a;
        }
        __syncthreads();
    }

    if (rg == 0) {
        const float4 nn = numS[threadIdx.x];
        const float4 dd = denS[threadIdx.x];
        float4 sv;
        sv.x = nn.x / dd.x;  sv.y = nn.y / dd.y;
        sv.z = nn.z / dd.z;  sv.w = nn.w / dd.w;
        *(float4*)(s + cb * OUT + og * 4) = sv;
        const float p2 = sv.x * sv.x + sv.y * sv.y + sv.z * sv.z + sv.w * sv.w;
        atomicAdd(n2, p2);
    }
}

// ---------------------------------------------------------------------------
// Kernel 3: global squash + Vsum update.  v = (n2/(1+n2)) * s / sqrt(n2).
// Single block; reads this iteration's n2 slot (no re-zeroing needed).
// ---------------------------------------------------------------------------
__global__ void caps_squash(const float* __restrict__ s,
                            float* __restrict__ Vsum,
                            const float* __restrict__ n2slot,
                            float* __restrict__ out,
                            int writeOut) {
    const float n2 = *n2slot;
    const float n = sqrtf(n2);
    const float factor = n2 / ((1.0f + n2) * n);   // scale / norm
    for (int i = threadIdx.x; i < SOUT; i += blockDim.x) {
        const float v = s[i] * factor;
        Vsum[i] += v;
        if (writeOut) out[i] = v;
    }
}

// ---------------------------------------------------------------------------
extern "C" void kernel_launch(void* const* d_in, const int* in_sizes, int n_in,
                              void* d_out, int out_size, void* d_ws, size_t ws_size,
                              hipStream_t stream) {
    const float* x = (const float*)d_in[0];
    const float* W = (const float*)d_in[1];
    float* out = (float*)d_out;

    // workspace layout: u (94.4 MB) | s | Vsum | n2[3]
    float* u    = (float*)d_ws;
    const long U = (long)CAPS * BATCH * RNODES * OUT;
    float* s    = u + U;
    float* Vsum = s + SOUT;
    float* n2   = Vsum + SOUT;

    caps_init<<<(SOUT + 255) / 256, 256, 0, stream>>>(Vsum, n2);

    // C*R = 11520 waves, 8 waves per 256-thread block
    caps_gemm<<<(CAPS * RNODES) / 8, 256, 0, stream>>>(x, W, u);

    for (int t = 0; t < NITER; ++t) {
        caps_route<<<CAPS * BATCH, 256, 0, stream>>>(u, Vsum, s, n2 + t);
        caps_squash<<<1, 256, 0, stream>>>(s, Vsum, n2 + t, out,
                                           (t == NITER - 1) ? 1 : 0);
    }
}